// TopkPointExtractor_14267881358076
// MI455X (gfx1250) — compile-verified
//
#include <hip/hip_runtime.h>
#include <hip/hip_bf16.h>

// ---------------- problem constants ----------------
#define BATCH   8
#define ENCO    64          // output channels of 1x1 conv
#define NBF     512         // input channels
#define HH      122
#define WWD     122
#define HWP     (HH * WWD)  // 14884 pixels per batch
#define FM_W    128
#define FM_HW   (128 * 128) // 16384, channel stride in featureMaps
#define KSEL    2048
#define NTILES  ((HWP + 15) / 16) // 931 16-pixel tiles per batch
#define PADV    1.0e-4f

typedef float v2f __attribute__((ext_vector_type(2)));
typedef float v8f __attribute__((ext_vector_type(8)));

// =====================================================================
// Kernel 1: pf[b,o,p] = sum_c W[o,c] * FM[b,c,h+3,w+3] + bias[o]
// One wave (32 lanes) computes a 64(o) x 16(pixel) tile via
// V_WMMA_F32_16X16X4_F32, looping K=512 in steps of 4.
// =====================================================================
__global__ void __launch_bounds__(256) pf_gemm_wmma(
    const float* __restrict__ fm, const float* __restrict__ cw,
    const float* __restrict__ cb, float* __restrict__ pf)
{
    const int lane = threadIdx.x & 31;
    const int l16  = lane & 15;
    const int half = lane >> 4;            // selects K-pair {0,1} vs {2,3}
    const int g    = blockIdx.x * 8 + (threadIdx.x >> 5);
    const int b    = g / NTILES;
    const int tile = g - b * NTILES;

    const int p  = tile * 16 + l16;        // flattened pixel (column of B/D)
    const int pc = p < HWP ? p : HWP - 1;  // clamp loads; stores predicated
    const int h  = pc / WWD;
    const int w  = pc - h * WWD;

    // B operand: f[c, pixel], c = 4k + 2*half + {0,1}
    const float* bptr = fm + (size_t)b * NBF * FM_HW
                           + (size_t)(2 * half) * FM_HW
                           + (size_t)(h + 3) * FM_W + (w + 3);
    // A operand: conv_w rows (o = 16*m + l16), K offset 4k + 2*half
    const float* a0 = cw + (size_t)l16 * NBF + 2 * half;
    const float* a1 = a0 + 16 * NBF;
    const float* a2 = a0 + 32 * NBF;
    const float* a3 = a0 + 48 * NBF;

    v8f acc0 = {}, acc1 = {}, acc2 = {}, acc3 = {};

    for (int k = 0; k < NBF / 4; ++k) {
        v2f Bv;
        Bv[0] = bptr[0];
        Bv[1] = bptr[FM_HW];
        // locality=3 -> WGP-scope prefetch: pull next K-step's lines into
        // all cache levels (loc=0 emitted SCOPE_SYS = GL2-only, wasted).
        __builtin_prefetch(bptr + 4 * (size_t)FM_HW, 0, 3);
        bptr += 4 * (size_t)FM_HW;

        v2f A0 = *(const v2f*)a0; a0 += 4;
        v2f A1 = *(const v2f*)a1; a1 += 4;
        v2f A2 = *(const v2f*)a2; a2 += 4;
        v2f A3 = *(const v2f*)a3; a3 += 4;

        acc0 = __builtin_amdgcn_wmma_f32_16x16x4_f32(false, A0, false, Bv,
                                                     (short)0, acc0, false, false);
        acc1 = __builtin_amdgcn_wmma_f32_16x16x4_f32(false, A1, false, Bv,
                                                     (short)0, acc1, false, false);
        acc2 = __builtin_amdgcn_wmma_f32_16x16x4_f32(false, A2, false, Bv,
                                                     (short)0, acc2, false, false);
        acc3 = __builtin_amdgcn_wmma_f32_16x16x4_f32(false, A3, false, Bv,
                                                     (short)0, acc3, false, false);
    }

    if (p < HWP) {
        float* out = pf + ((size_t)b * ENCO) * HWP + p;
        #pragma unroll
        for (int r = 0; r < 8; ++r) {
            const int m = r + 8 * half;  // D row for this lane/element
            out[(size_t)(m)      * HWP] = acc0[r] + cb[m];
            out[(size_t)(m + 16) * HWP] = acc1[r] + cb[m + 16];
            out[(size_t)(m + 32) * HWP] = acc2[r] + cb[m + 32];
            out[(size_t)(m + 48) * HWP] = acc3[r] + cb[m + 48];
        }
    }
}

// =====================================================================
// Kernel 2a: per-pixel L2 norm over the 64 channels of pf
// =====================================================================
__global__ void norms_kernel(const float* __restrict__ pf, float* __restrict__ nrm)
{
    int t = blockIdx.x * blockDim.x + threadIdx.x;
    if (t >= BATCH * HWP) return;
    int b = t / HWP;
    int p = t - b * HWP;
    const float* base = pf + ((size_t)b * ENCO) * HWP + p;
    float s = 0.f;
    #pragma unroll 8
    for (int c = 0; c < ENCO; ++c) {
        float v = base[(size_t)c * HWP];
        s += v * v;
    }
    nrm[t] = sqrtf(s);
}

// =====================================================================
// Kernel 2b: x = -(cos_top + cos_bot + cos_left + cos_right)/4
// top: neighbor w+1, bot: w-1, left: h-1, right: h+1 (PAD boundary)
// =====================================================================
__global__ void cosx_kernel(const float* __restrict__ pf,
                            const float* __restrict__ nrm,
                            float* __restrict__ xv)
{
    int t = blockIdx.x * blockDim.x + threadIdx.x;
    if (t >= BATCH * HWP) return;
    int b = t / HWP;
    int p = t - b * HWP;
    int h = p / WWD;
    int w = p - h * WWD;

    const bool vt = (w + 1) < WWD;  // top   -> +1
    const bool vb = w > 0;          // bot   -> -1
    const bool vl = h > 0;          // left  -> -122
    const bool vr = (h + 1) < HH;   // right -> +122

    const int ot = vt ? 1 : 0;
    const int ob = vb ? -1 : 0;
    const int ol = vl ? -WWD : 0;
    const int orr = vr ? WWD : 0;

    const float* base = pf + ((size_t)b * ENCO) * HWP + p;
    float dt = 0.f, db = 0.f, dl = 0.f, dr = 0.f, sumf = 0.f;
    #pragma unroll 4
    for (int c = 0; c < ENCO; ++c) {
        const size_t off = (size_t)c * HWP;
        float f = base[off];
        sumf += f;
        dt += f * base[off + ot];
        db += f * base[off + ob];
        dl += f * base[off + ol];
        dr += f * base[off + orr];
    }

    const float np = nrm[t];
    const float edge = PADV * sumf / (8.f * np); // PAD^2*sumf / (np * 8*PAD)
    const float ct = vt ? dt / (np * nrm[t + 1])   : edge;
    const float cbt = vb ? db / (np * nrm[t - 1])  : edge;
    const float cl = vl ? dl / (np * nrm[t - WWD]) : edge;
    const float cr = vr ? dr / (np * nrm[t + WWD]) : edge;

    xv[t] = -(ct + cbt + cl + cr) * 0.25f;
}

// =====================================================================
// Kernel 3: exact per-batch top-K (descending, ties -> smaller index).
// 4-pass radix threshold in LDS, compaction, 2048-elem bitonic sort.
// =====================================================================
__device__ __forceinline__ unsigned sortable(float f)
{
    unsigned u = __float_as_uint(f);
    return (u & 0x80000000u) ? ~u : (u | 0x80000000u);
}

__device__ void bitonic_desc(unsigned long long* buf, unsigned n, int tid, int nt)
{
    for (unsigned k = 2; k <= n; k <<= 1) {
        for (unsigned j = k >> 1; j > 0; j >>= 1) {
            __syncthreads();
            for (unsigned i = (unsigned)tid; i < n; i += (unsigned)nt) {
                unsigned ixj = i ^ j;
                if (ixj > i) {
                    unsigned long long a = buf[i], c = buf[ixj];
                    bool up = ((i & k) == 0);
                    if (up ? (a < c) : (a > c)) { buf[i] = c; buf[ixj] = a; }
                }
            }
        }
    }
    __syncthreads();
}

__global__ void __launch_bounds__(1024) topk_kernel(const float* __restrict__ xv,
                                                    int* __restrict__ out_idx)
{
    __shared__ unsigned int hist[256];
    __shared__ unsigned long long bufG[KSEL];
    __shared__ unsigned long long bufE[KSEL];
    __shared__ unsigned int sPrefix, sNeed, sCntG, sCntE;

    const int b   = blockIdx.x;
    const int tid = threadIdx.x;
    const int nt  = blockDim.x;
    const float* x = xv + (size_t)b * HWP;

    if (tid == 0) { sPrefix = 0u; sNeed = KSEL; }
    __syncthreads();

    // ---- 4-pass radix refinement of the K-th largest key ----
    for (int shift = 24; shift >= 0; shift -= 8) {
        for (int i = tid; i < 256; i += nt) hist[i] = 0u;
        __syncthreads();
        const unsigned pmask = (shift == 24) ? 0u : (0xFFFFFFFFu << (shift + 8));
        const unsigned pref  = sPrefix;
        for (int i = tid; i < HWP; i += nt) {
            unsigned s = sortable(x[i]);
            if ((s & pmask) == pref) atomicAdd(&hist[(s >> shift) & 0xFF], 1u);
        }
        __syncthreads();
        if (tid == 0) {
            unsigned need = sNeed, cum = 0u;
            int bsel = 0;
            for (int bb = 255; bb >= 0; --bb) {
                if (cum + hist[bb] >= need) { bsel = bb; break; }
                cum += hist[bb];
            }
            sPrefix |= ((unsigned)bsel << shift);
            sNeed = need - cum;
        }
        __syncthreads();
    }

    const unsigned thr = sPrefix;
    if (tid == 0) { sCntG = 0u; sCntE = 0u; }
    __syncthreads();

    // ---- compaction:  > thr into bufG,  == thr into bufE ----
    for (int i = tid; i < HWP; i += nt) {
        unsigned s = sortable(x[i]);
        unsigned long long key =
            ((unsigned long long)s << 32) |
            (unsigned long long)(0xFFFFFFFFu - (unsigned)i); // tie: smaller idx wins
        if (s > thr) {
            unsigned q = atomicAdd(&sCntG, 1u);
            bufG[q] = key;
        } else if (s == thr) {
            unsigned q = atomicAdd(&sCntE, 1u);
            if (q < KSEL) bufE[q] = key;
        }
    }
    __syncthreads();

    const unsigned cntG = sCntG;
    const unsigned need = sNeed;                       // equals to take
    const unsigned cntE = sCntE < (unsigned)KSEL ? sCntE : (unsigned)KSEL;

    unsigned nE = 1;
    while (nE < cntE) nE <<= 1;
    for (unsigned i = cntE + tid; i < nE; i += nt) bufE[i] = 0ull;
    __syncthreads();
    bitonic_desc(bufE, nE, tid, nt);                   // smallest-index equals first

    for (unsigned i = (unsigned)tid; i < need; i += (unsigned)nt)
        bufG[cntG + i] = bufE[i];                      // cntG + need == KSEL
    __syncthreads();
    bitonic_desc(bufG, KSEL, tid, nt);                 // final descending order

    for (int i = tid; i < KSEL; i += nt)
        out_idx[(size_t)b * KSEL + i] =
            (int)(0xFFFFFFFFu - (unsigned)(bufG[i] & 0xFFFFFFFFull));
}

// =====================================================================
// Kernel 4: gather  out[b,k,o] = pf[b,o,idx[b,k]]
// =====================================================================
__global__ void gather_kernel(const float* __restrict__ pf,
                              const int* __restrict__ idx,
                              float* __restrict__ out)
{
    int t = blockIdx.x * blockDim.x + threadIdx.x;   // (b,k,o) flattened, 2^20
    int o = t & (ENCO - 1);
    int k = (t >> 6) & (KSEL - 1);
    int b = t >> 17;
    int p = idx[(size_t)b * KSEL + k];
    out[t] = pf[((size_t)b * ENCO + o) * HWP + p];
}

// =====================================================================
// host launcher
// =====================================================================
extern "C" void kernel_launch(void* const* d_in, const int* in_sizes, int n_in,
                              void* d_out, int out_size, void* d_ws, size_t ws_size,
                              hipStream_t stream)
{
    const float* fm = (const float*)d_in[0]; // (8, 512, 128, 128)
    const float* cw = (const float*)d_in[1]; // (64, 512)
    const float* cb = (const float*)d_in[2]; // (64,)

    // workspace layout (~31.5 MB total)
    float* pf  = (float*)d_ws;                             // 8*64*14884 f32
    float* nrm = pf  + (size_t)BATCH * ENCO * HWP;         // 8*14884 f32
    float* xv  = nrm + (size_t)BATCH * HWP;                // 8*14884 f32
    int*   idx = (int*)(xv + (size_t)BATCH * HWP);         // 8*2048 i32
    float* out = (float*)d_out;                            // (8, 2048, 64) f32

    pf_gemm_wmma<<<NTILES, 256, 0, stream>>>(fm, cw, cb, pf);

    const int n2 = BATCH * HWP;
    norms_kernel<<<(n2 + 255) / 256, 256, 0, stream>>>(pf, nrm);
    cosx_kernel <<<(n2 + 255) / 256, 256, 0, stream>>>(pf, nrm, xv);

    topk_kernel<<<BATCH, 1024, 0, stream>>>(xv, idx);

    gather_kernel<<<(BATCH * KSEL * ENCO) / 256, 256, 0, stream>>>(pf, idx, out);
}